// GATEncoder_63256278335600
// MI455X (gfx1250) — compile-verified
//
#include <hip/hip_runtime.h>

#define NEG_SLOPE 0.2f
#define BN_EPS 1e-5f

typedef float v2f __attribute__((ext_vector_type(2)));
typedef float v8f __attribute__((ext_vector_type(8)));

// ---------------------------------------------------------------------------
// Monotonic float<->uint mapping so we can use atomicMax(u32) as float max.
// ---------------------------------------------------------------------------
__device__ __forceinline__ unsigned fenc(float f) {
    unsigned u = __float_as_uint(f);
    return (u & 0x80000000u) ? ~u : (u | 0x80000000u);
}
__device__ __forceinline__ float fdec(unsigned u) {
    return __uint_as_float((u & 0x80000000u) ? (u & 0x7fffffffu) : ~u);
}

// ---------------------------------------------------------------------------
// FP32 WMMA: D = A[M,K] x B[K,N]  (row-major, M%16==0, N%16==0, K%4==0)
// V_WMMA_F32_16X16X4_F32; one wave computes one 16x16 tile.
// B (<= 32KB) is staged block-wide into LDS via the gfx1250 async-DMA path
// (global_load_async_to_lds_b128, drained with s_wait_asynccnt), then the
// K-loop feeds WMMA from ds_load instead of per-iteration global loads.
// A-frag (16x4 f32, 2 VGPR): lanes 0-15 hold K=0,1; lanes 16-31 hold K=2,3.
// C/D (16x16 f32, 8 VGPR): lanes 0-15 -> rows 0..7, lanes 16-31 -> rows 8..15.
// ---------------------------------------------------------------------------
__device__ __forceinline__ v8f wmma4(v2f a, v2f b, v8f c) {
    return __builtin_amdgcn_wmma_f32_16x16x4_f32(
        /*neg_a=*/false, a, /*neg_b=*/false, b,
        /*c_mod=*/(short)0, c, /*reuse_a=*/false, /*reuse_b=*/false);
}

#define BS_MAX 8192   // floats: enough for W1 (16x256) and W2 (256x32)

__global__ void gemm_wmma_f32(const float* __restrict__ A, const float* __restrict__ B,
                              float* __restrict__ D, int M, int N, int K) {
    __shared__ float Bs[BS_MAX];
    // Low 32 bits of a generic pointer into the LDS aperture are the LDS byte
    // offset (ISA 10.2).
    unsigned bs_base = (unsigned)(size_t)(void*)Bs;

    // --- stage B[K,N] into LDS with async 128-bit DMA loads (all threads) ---
    int tid = (int)threadIdx.x;
    int nquads = (K * N) >> 2;                 // 16B chunks
    for (int i = tid; i < nquads; i += (int)blockDim.x) {
        asm volatile("global_load_async_to_lds_b128 %0, %1, off"
                     :: "v"(bs_base + (unsigned)i * 16u),
                        "v"(B + (size_t)i * 4)
                     : "memory");
    }
    asm volatile("s_wait_asynccnt 0x0" ::: "memory");
    __syncthreads();

    int wave = (int)blockIdx.x * ((int)blockDim.x >> 5) + (tid >> 5);
    int lane = tid & 31;
    int ntiles = N >> 4;
    int mtiles = (M + 15) >> 4;
    if (wave >= mtiles * ntiles) return;       // after the barrier: safe
    int mt = wave / ntiles, nt = wave % ntiles;
    int half = lane >> 4, l16 = lane & 15;

    int arow = mt * 16 + l16;                  // A rows of this tile
    int bcol = nt * 16 + l16;                  // B cols of this tile
    const float* Ap = A + (size_t)arow * K;

    v8f acc = {0.f, 0.f, 0.f, 0.f, 0.f, 0.f, 0.f, 0.f};
    for (int k = 0; k < K; k += 4) {
        int ka = k + half * 2;
        v2f a, b;
        a.x = Ap[ka];
        a.y = Ap[ka + 1];
        b.x = Bs[ka * N + bcol];
        b.y = Bs[(ka + 1) * N + bcol];
        if (k + 4 < K) __builtin_prefetch(&Ap[k + 4], 0, 0);
        acc = wmma4(a, b, acc);
    }
    int row0 = mt * 16 + half * 8;
    int col  = nt * 16 + l16;
#pragma unroll
    for (int r = 0; r < 8; ++r)
        D[(size_t)(row0 + r) * N + col] = acc[r];
}

// ---------------------------------------------------------------------------
// Utility: zero a float/uint region (harness poisons ws with 0xAA each run).
// ---------------------------------------------------------------------------
__global__ void zero_f32(float* p, long n) {
    long i = blockIdx.x * (long)blockDim.x + threadIdx.x;
    if (i < n) p[i] = 0.f;
}

// ---------------------------------------------------------------------------
// Per-node attention logits: as[n,h] = <h[n,h,:], a_src[h,:]>, same for dst.
// ---------------------------------------------------------------------------
__global__ void node_alpha(const float* __restrict__ h, const float* __restrict__ a_src,
                           const float* __restrict__ a_dst, float* __restrict__ as,
                           float* __restrict__ ad, int Nn, int H, int Dd) {
    int n = blockIdx.x * blockDim.x + threadIdx.x;
    if (n >= Nn) return;
    int C = H * Dd;
    for (int hh = 0; hh < H; ++hh) {
        float s = 0.f, d = 0.f;
        const float* hp  = h + (size_t)n * C + hh * Dd;
        const float* asp = a_src + hh * Dd;
        const float* adp = a_dst + hh * Dd;
        for (int c = 0; c < Dd; ++c) { s += hp[c] * asp[c]; d += hp[c] * adp[c]; }
        as[(size_t)n * H + hh] = s;
        ad[(size_t)n * H + hh] = d;
    }
}

// Edge list: e < E -> (ei[e], ei[E+e]); e >= E -> self loop (e-E, e-E).
__device__ __forceinline__ void edge_sd(const int* ei, int E, long e, int& s, int& d) {
    if (e < E) { s = ei[e]; d = ei[E + e]; } else { s = d = (int)(e - E); }
}

// Phase 1: e = leaky_relu(as[src]+ad[dst]); segment-max into menc[dst] (ordered u32).
__global__ void edge_logits(const int* __restrict__ ei, int E, int Nn,
                            const float* __restrict__ as, const float* __restrict__ ad,
                            float* __restrict__ el, unsigned* __restrict__ menc, int H) {
    long e = blockIdx.x * (long)blockDim.x + threadIdx.x;
    if (e >= (long)E + Nn) return;
    int s, d; edge_sd(ei, E, e, s, d);
    for (int hh = 0; hh < H; ++hh) {
        float v = as[(size_t)s * H + hh] + ad[(size_t)d * H + hh];
        v = v > 0.f ? v : NEG_SLOPE * v;
        el[(size_t)e * H + hh] = v;
        atomicMax(&menc[(size_t)d * H + hh], fenc(v));
    }
}

// Phase 2: w = exp(e - max[dst]); segment-sum into z[dst].
__global__ void edge_expsum(const int* __restrict__ ei, int E, int Nn,
                            const float* __restrict__ el, const unsigned* __restrict__ menc,
                            float* __restrict__ w, float* __restrict__ z, int H) {
    long e = blockIdx.x * (long)blockDim.x + threadIdx.x;
    if (e >= (long)E + Nn) return;
    int s, d; edge_sd(ei, E, e, s, d);
    (void)s;
    for (int hh = 0; hh < H; ++hh) {
        float v = __expf(el[(size_t)e * H + hh] - fdec(menc[(size_t)d * H + hh]));
        w[(size_t)e * H + hh] = v;
        atomicAdd(&z[(size_t)d * H + hh], v);
    }
}

// Phase 3: out[dst,:] += (w/z[dst]) * hfeat[src,:]  (float4 loads, f32 atomics).
__global__ void edge_agg(const int* __restrict__ ei, int E, int Nn,
                         const float* __restrict__ hfeat, const float* __restrict__ w,
                         const float* __restrict__ z, float* __restrict__ out, int C, int H) {
    long tid = blockIdx.x * (long)blockDim.x + threadIdx.x;
    int quads = C >> 2;
    long total = ((long)E + Nn) * quads;
    if (tid >= total) return;
    int q = (int)(tid % quads);
    long e = tid / quads;
    int s, d; edge_sd(ei, E, e, s, d);
    int c  = q * 4;
    int Dd = C / H;
    int hh = c / Dd;
    float alpha = w[(size_t)e * H + hh] / z[(size_t)d * H + hh];
    const float4 hv = *(const float4*)(hfeat + (size_t)s * C + c);
    float* op = out + (size_t)d * C + c;
    atomicAdd(op + 0, alpha * hv.x);
    atomicAdd(op + 1, alpha * hv.y);
    atomicAdd(op + 2, alpha * hv.z);
    atomicAdd(op + 3, alpha * hv.w);
}

// ---------------------------------------------------------------------------
// BatchNorm stats: one block per channel, LDS tree reduction.
// var is biased (ddof=0), matching jnp.var.
// ---------------------------------------------------------------------------
__global__ void bn_stats(const float* __restrict__ x, int Nn, int C,
                         float* __restrict__ mean, float* __restrict__ var) {
    int c = blockIdx.x;
    float s = 0.f, s2 = 0.f;
    for (int n = threadIdx.x; n < Nn; n += blockDim.x) {
        float v = x[(size_t)n * C + c];
        s += v; s2 += v * v;
    }
    __shared__ float ss[256], sq[256];
    ss[threadIdx.x] = s; sq[threadIdx.x] = s2;
    __syncthreads();
    for (int st = blockDim.x >> 1; st > 0; st >>= 1) {
        if ((int)threadIdx.x < st) {
            ss[threadIdx.x] += ss[threadIdx.x + st];
            sq[threadIdx.x] += sq[threadIdx.x + st];
        }
        __syncthreads();
    }
    if (threadIdx.x == 0) {
        float m = ss[0] / (float)Nn;
        mean[c] = m;
        var[c]  = sq[0] / (float)Nn - m * m;
    }
}

// Fused BN + ELU. (Pre-BN bias terms cancel inside BN and are skipped.)
__global__ void bn_elu(const float* __restrict__ x, const float* __restrict__ mean,
                       const float* __restrict__ var, const float* __restrict__ g,
                       const float* __restrict__ be, float* __restrict__ y,
                       long total, int C) {
    long i = blockIdx.x * (long)blockDim.x + threadIdx.x;
    if (i >= total) return;
    int c = (int)(i % C);
    float t = (x[i] - mean[c]) * rsqrtf(var[c] + BN_EPS) * g[c] + be[c];
    y[i] = t > 0.f ? t : __expf(t) - 1.f;
}

// ---------------------------------------------------------------------------
// Launch
// ---------------------------------------------------------------------------
static inline long cdiv(long a, long b) { return (a + b - 1) / b; }

extern "C" void kernel_launch(void* const* d_in, const int* in_sizes, int n_in,
                              void* d_out, int out_size, void* d_ws, size_t ws_size,
                              hipStream_t stream) {
    const float* x      = (const float*)d_in[0];
    const int*   ei     = (const int*)  d_in[1];
    const float* W1     = (const float*)d_in[2];
    const float* a_src1 = (const float*)d_in[3];
    const float* a_dst1 = (const float*)d_in[4];
    const float* g1     = (const float*)d_in[6];
    const float* be1    = (const float*)d_in[7];
    const float* W2     = (const float*)d_in[8];
    const float* a_src2 = (const float*)d_in[9];
    const float* a_dst2 = (const float*)d_in[10];
    const float* g2     = (const float*)d_in[12];
    const float* be2    = (const float*)d_in[13];

    const int Nn = in_sizes[0] / 16;   // 20000 nodes
    const int E  = in_sizes[1] / 2;    // 320000 edges
    const long EE = (long)E + Nn;      // + self loops

    // ---- workspace layout (floats) ----
    float* ws = (float*)d_ws;
    size_t o = 0;
    float*    h1    = ws + o; o += (size_t)Nn * 256;   // x @ W1
    float*    as1   = ws + o; o += (size_t)Nn * 4;
    float*    ad1   = ws + o; o += (size_t)Nn * 4;
    unsigned* menc1 = (unsigned*)(ws + o); o += (size_t)Nn * 4;
    float*    z1    = ws + o; o += (size_t)Nn * 4;
    float*    el1   = ws + o; o += (size_t)EE * 4;
    float*    w1    = ws + o; o += (size_t)EE * 4;
    float*    agg1  = ws + o; o += (size_t)Nn * 256;   // aggregated, then BN+ELU in place
    float*    mean1 = ws + o; o += 256;
    float*    var1  = ws + o; o += 256;
    float*    h2    = ws + o; o += (size_t)Nn * 32;    // agg1 @ W2
    float*    as2   = ws + o; o += (size_t)Nn;
    float*    ad2   = ws + o; o += (size_t)Nn;
    unsigned* menc2 = (unsigned*)(ws + o); o += (size_t)Nn;
    float*    z2    = ws + o; o += (size_t)Nn;
    float*    el2   = ws + o; o += (size_t)EE;
    float*    w2e   = ws + o; o += (size_t)EE;
    float*    agg2  = ws + o; o += (size_t)Nn * 32;
    float*    mean2 = ws + o; o += 32;
    float*    var2  = ws + o; o += 32;

    const int B = 256;

    // ---- zero accumulators (ws is poisoned between runs) ----
    zero_f32<<<cdiv((long)Nn * 4,   B), B, 0, stream>>>((float*)menc1, (long)Nn * 4);
    zero_f32<<<cdiv((long)Nn * 4,   B), B, 0, stream>>>(z1,            (long)Nn * 4);
    zero_f32<<<cdiv((long)Nn * 256, B), B, 0, stream>>>(agg1,          (long)Nn * 256);
    zero_f32<<<cdiv((long)Nn,       B), B, 0, stream>>>((float*)menc2, (long)Nn);
    zero_f32<<<cdiv((long)Nn,       B), B, 0, stream>>>(z2,            (long)Nn);
    zero_f32<<<cdiv((long)Nn * 32,  B), B, 0, stream>>>(agg2,          (long)Nn * 32);

    // ---- layer 1: GAT(16 -> 4 heads x 64, concat) ----
    {
        long waves = (long)cdiv(Nn, 16) * (256 / 16);
        gemm_wmma_f32<<<cdiv(waves * 32, B), B, 0, stream>>>(x, W1, h1, Nn, 256, 16);
    }
    node_alpha <<<cdiv(Nn, B), B, 0, stream>>>(h1, a_src1, a_dst1, as1, ad1, Nn, 4, 64);
    edge_logits<<<cdiv(EE, B), B, 0, stream>>>(ei, E, Nn, as1, ad1, el1, menc1, 4);
    edge_expsum<<<cdiv(EE, B), B, 0, stream>>>(ei, E, Nn, el1, menc1, w1, z1, 4);
    edge_agg   <<<cdiv(EE * 64, B), B, 0, stream>>>(ei, E, Nn, h1, w1, z1, agg1, 256, 4);
    bn_stats   <<<256, B, 0, stream>>>(agg1, Nn, 256, mean1, var1);
    bn_elu     <<<cdiv((long)Nn * 256, B), B, 0, stream>>>(agg1, mean1, var1, g1, be1,
                                                           agg1, (long)Nn * 256, 256);

    // ---- layer 2: GAT(256 -> 1 head x 32, mean) ----
    {
        long waves = (long)cdiv(Nn, 16) * (32 / 16);
        gemm_wmma_f32<<<cdiv(waves * 32, B), B, 0, stream>>>(agg1, W2, h2, Nn, 32, 256);
    }
    node_alpha <<<cdiv(Nn, B), B, 0, stream>>>(h2, a_src2, a_dst2, as2, ad2, Nn, 1, 32);
    edge_logits<<<cdiv(EE, B), B, 0, stream>>>(ei, E, Nn, as2, ad2, el2, menc2, 1);
    edge_expsum<<<cdiv(EE, B), B, 0, stream>>>(ei, E, Nn, el2, menc2, w2e, z2, 1);
    edge_agg   <<<cdiv(EE * 8, B), B, 0, stream>>>(ei, E, Nn, h2, w2e, z2, agg2, 32, 1);
    bn_stats   <<<32, B, 0, stream>>>(agg2, Nn, 32, mean2, var2);
    bn_elu     <<<cdiv((long)Nn * 32, B), B, 0, stream>>>(agg2, mean2, var2, g2, be2,
                                                          (float*)d_out, (long)Nn * 32, 32);
}